// Encoder_29180007809406
// MI455X (gfx1250) — compile-verified
//
#include <hip/hip_runtime.h>

typedef float v2f __attribute__((ext_vector_type(2)));
typedef float v8f __attribute__((ext_vector_type(8)));

// ---------------------------------------------------------------------------
// Degree / normalization kernels
// ---------------------------------------------------------------------------
__global__ void k_init_deg(float* deg, int n) {
    int i = blockIdx.x * blockDim.x + threadIdx.x;
    if (i < n) deg[i] = 1.0f;   // self-loop contributes +1
}

__global__ void k_count_deg(const int* __restrict__ dst, float* deg, int e) {
    int i = blockIdx.x * blockDim.x + threadIdx.x;
    if (i < e) atomicAdd(&deg[dst[i]], 1.0f);
}

__global__ void k_rsqrt(float* dis, int n) {
    int i = blockIdx.x * blockDim.x + threadIdx.x;
    if (i < n) dis[i] = rsqrtf(dis[i]);
}

// ---------------------------------------------------------------------------
// GEMM: H[n x C] = A[n x 64] * W[64 x C], C in {32,64}, n % 16 == 0.
// One wave computes one 16x16 output tile via 16 x V_WMMA_F32_16X16X4_F32.
// blockDim.x = 32 * (C/16); grid.x = n/16.  C and LDA are compile-time so all
// loop loads use immediate offsets (no per-iteration 64-bit address math).
//
// f32 16x16x4 VGPR layouts (ISA 7.12.2):
//   A (16x4): lane m = lane%16, VGPR{0,1} hold K = 2*(lane/16) + {0,1}
//   B (4x16): lane n = lane%16, VGPR{0,1} hold K = 2*(lane/16) + {0,1}
//   C/D (16x16, 8 VGPRs): VGPR v, lane l -> M = v + 8*(l/16), N = l%16
// ---------------------------------------------------------------------------
template <int C, int LDA>
__global__ void k_gemm_wmma(const float* __restrict__ A,
                            const float* __restrict__ W,
                            float* __restrict__ H) {
    const int lane  = threadIdx.x & 31;
    const int wave  = threadIdx.x >> 5;
    const int m     = lane & 15;        // A row within tile / B-D column within tile
    const int half  = lane >> 4;
    const int kbase = half * 2;
    const int row0  = blockIdx.x * 16;
    const int col   = wave * 16 + m;    // output column handled by this lane

    // per-lane bases; all subsequent accesses use compile-time immediate offsets
    const float* __restrict__ abase = A + (size_t)(row0 + m) * LDA + kbase;
    const float* __restrict__ wbase = W + kbase * C + col;

    v8f acc = {0.f, 0.f, 0.f, 0.f, 0.f, 0.f, 0.f, 0.f};

#pragma unroll
    for (int k = 0; k < 64; k += 4) {
        v2f a;                                   // contiguous pair -> global_load_b64
        a.x = abase[k];
        a.y = abase[k + 1];
        v2f b;
        b.x = wbase[k * C];                      // constant offsets
        b.y = wbase[(k + 1) * C];
        acc = __builtin_amdgcn_wmma_f32_16x16x4_f32(
            /*neg_a=*/false, a, /*neg_b=*/false, b,
            /*c_mod=*/(short)0, acc, /*reuse_a=*/false, /*reuse_b=*/false);
    }

    float* __restrict__ hbase = H + (size_t)(row0 + half * 8) * C + col;
#pragma unroll
    for (int v = 0; v < 8; ++v) {
        hbase[v * C] = acc[v];
    }
}

// ---------------------------------------------------------------------------
// agg[i] = h[i] * dis[i]^2  (self-loop term), float4 per thread
// ---------------------------------------------------------------------------
template <int C>
__global__ void k_selfloop(const float* __restrict__ h, const float* __restrict__ dis,
                           float* __restrict__ agg, int n) {
    constexpr int CPE = C >> 2;
    int i = blockIdx.x * blockDim.x + threadIdx.x;
    if (i >= n * CPE) return;
    int node = i / CPE;                // shift (CPE power of two)
    int c4   = (i % CPE) << 2;
    float w = dis[node];
    w *= w;
    const float4 v = *reinterpret_cast<const float4*>(h + (size_t)node * C + c4);
    float4* o = reinterpret_cast<float4*>(agg + (size_t)node * C + c4);
    *o = make_float4(v.x * w, v.y * w, v.z * w, v.w * w);
}

// ---------------------------------------------------------------------------
// Edge scatter: agg[dst] += h[src] * dis[src]*dis[dst].  C/4 threads per edge.
// ---------------------------------------------------------------------------
template <int C>
__global__ void k_scatter(const float* __restrict__ h,
                          const int* __restrict__ src, const int* __restrict__ dst,
                          const float* __restrict__ dis,
                          float* __restrict__ agg, int e) {
    constexpr int CPE = C >> 2;
    int i = blockIdx.x * blockDim.x + threadIdx.x;
    if (i >= e * CPE) return;
    int ed = i / CPE;                  // shift
    int c4 = (i % CPE) << 2;
    int s = src[ed];
    int d = dst[ed];
    float coef = dis[s] * dis[d];
    const float4 v = *reinterpret_cast<const float4*>(h + (size_t)s * C + c4);
    float* o = agg + (size_t)d * C + c4;
    atomicAdd(o + 0, v.x * coef);
    atomicAdd(o + 1, v.y * coef);
    atomicAdd(o + 2, v.z * coef);
    atomicAdd(o + 3, v.w * coef);
}

// ---------------------------------------------------------------------------
// out[node*160 + c] = (RELU ? max(0, agg+b) : agg+b) — writes into concat slice
// ---------------------------------------------------------------------------
template <int C, int RELU>
__global__ void k_finalize(const float* __restrict__ agg, const float* __restrict__ bias,
                           float* __restrict__ out, int n) {
    int i = blockIdx.x * blockDim.x + threadIdx.x;
    if (i >= n * C) return;
    int node = i / C;                  // shift
    int c    = i % C;
    float v = agg[i] + bias[c];
    if (RELU) v = fmaxf(v, 0.0f);
    out[(size_t)node * 160 + c] = v;
}

// ---------------------------------------------------------------------------
extern "C" void kernel_launch(void* const* d_in, const int* in_sizes, int n_in,
                              void* d_out, int out_size, void* d_ws, size_t ws_size,
                              hipStream_t stream) {
    const float* x  = (const float*)d_in[0];
    const int*   ei = (const int*)  d_in[1];
    const float* W1 = (const float*)d_in[2];
    const float* b1 = (const float*)d_in[3];
    const float* W2 = (const float*)d_in[4];
    const float* b2 = (const float*)d_in[5];
    const float* W3 = (const float*)d_in[6];
    const float* b3 = (const float*)d_in[7];
    float* out = (float*)d_out;

    const int N = in_sizes[0] / 64;   // 100000 (divisible by 16)
    const int E = in_sizes[1] / 2;    // 1600000
    const int* src = ei;
    const int* dst = ei + E;

    // workspace: dis[N] | h[N*64] | agg[N*64]
    float* dis = (float*)d_ws;
    float* h   = dis + N;
    float* agg = h + (size_t)N * 64;

    const int TB = 256;

    // degree -> deg^{-1/2}
    k_init_deg <<<(N + TB - 1) / TB, TB, 0, stream>>>(dis, N);
    k_count_deg<<<(E + TB - 1) / TB, TB, 0, stream>>>(dst, dis, E);
    k_rsqrt    <<<(N + TB - 1) / TB, TB, 0, stream>>>(dis, N);

    // ---- layer 1: x[N,64] @ W1[64,64] -> x1 (relu) -> out cols 96..159
    k_gemm_wmma<64, 64><<<N / 16, 128, 0, stream>>>(x, W1, h);
    k_selfloop<64><<<(N * 16 + TB - 1) / TB, TB, 0, stream>>>(h, dis, agg, N);
    k_scatter <64><<<(E * 16 + TB - 1) / TB, TB, 0, stream>>>(h, src, dst, dis, agg, E);
    k_finalize<64, 1><<<(N * 64 + TB - 1) / TB, TB, 0, stream>>>(agg, b1, out + 96, N);

    // ---- layer 2: x1 (out+96, lda=160) @ W2[64,64] -> x2 (relu) -> out cols 32..95
    k_gemm_wmma<64, 160><<<N / 16, 128, 0, stream>>>(out + 96, W2, h);
    k_selfloop<64><<<(N * 16 + TB - 1) / TB, TB, 0, stream>>>(h, dis, agg, N);
    k_scatter <64><<<(E * 16 + TB - 1) / TB, TB, 0, stream>>>(h, src, dst, dis, agg, E);
    k_finalize<64, 1><<<(N * 64 + TB - 1) / TB, TB, 0, stream>>>(agg, b2, out + 32, N);

    // ---- layer 3: x2 (out+32, lda=160) @ W3[64,32] -> x3 (no relu) -> out cols 0..31
    k_gemm_wmma<32, 160><<<N / 16, 64, 0, stream>>>(out + 32, W3, h);
    k_selfloop<32><<<(N * 8 + TB - 1) / TB, TB, 0, stream>>>(h, dis, agg, N);
    k_scatter <32><<<(E * 8 + TB - 1) / TB, TB, 0, stream>>>(h, src, dst, dis, agg, E);
    k_finalize<32, 0><<<(N * 32 + TB - 1) / TB, TB, 0, stream>>>(agg, b3, out + 0, N);
}